// VariantDifferencePooling_87136296501256
// MI455X (gfx1250) — compile-verified
//
#include <hip/hip_runtime.h>

// Problem constants (match reference)
#define B_    32
#define L_    4096
#define D_    768
#define W_    32
#define WLEN  65      // 2*W+1 window length
#define NKB   17      // ceil(65/4) K-blocks of the 16x16x4 f32 WMMA
#define LN_EPS 1e-5f

typedef float v2f __attribute__((ext_vector_type(2)));
typedef float v8f __attribute__((ext_vector_type(8)));

// One workgroup per batch element. 256 threads = 8 waves (wave32).
// Each wave computes 6 channel-tiles of 16 via V_WMMA_F32_16X16X4_F32:
//   D[16x16] += A[16x4] * B[4x16]
// A rows are all identical = window weights (valid/count or 0), so every
// row of D equals pooled[tile] — the masked window mean, in full f32.
__global__ __launch_bounds__(256) void vdp_wmma_kernel(
    const float* __restrict__ ref_repr,
    const float* __restrict__ alt_repr,
    const int*   __restrict__ variant_pos,
    const float* __restrict__ gamma,
    const float* __restrict__ beta,
    float*       __restrict__ out)
{
    __shared__ float s_pooled[D_];
    __shared__ float s_part[16];   // [0..7] sums, [8..15] sum-squares per wave

    const int b    = blockIdx.x;
    const int tid  = threadIdx.x;
    const int wave = tid >> 5;
    const int lane = tid & 31;
    const int half = lane >> 4;    // 0: lanes 0-15, 1: lanes 16-31
    const int col  = lane & 15;    // N (channel within tile)

    const int pos = variant_pos[b];
    const int lo  = max(pos - W_, 0);
    const int hi  = min(pos + W_, L_ - 1);
    const float inv_count = 1.0f / (float)(hi - lo + 1);

    const float* __restrict__ refb = ref_repr + (size_t)b * L_ * D_;
    const float* __restrict__ altb = alt_repr + (size_t)b * L_ * D_;

    // ---- pooled window mean via f32 WMMA ----
    for (int t = 0; t < 6; ++t) {
        const int n0 = (wave * 6 + t) * 16;  // channel tile base
        v8f acc = {};
        for (int kb = 0; kb < NKB; ++kb) {
            // ISA 32-bit 16x4 A layout: VGPR0 -> K = 4kb + 2*half,
            //                           VGPR1 -> K = 4kb + 2*half + 1
            const int j0 = kb * 4 + half * 2;
            const int j1 = j0 + 1;
            const int idx0 = pos - W_ + j0;
            const int idx1 = pos - W_ + j1;

            // A-matrix: weight row, identical across all 16 M-rows.
            v2f a;
            a.x = (j0 < WLEN && idx0 >= 0 && idx0 < L_) ? inv_count : 0.0f;
            a.y = (j1 < WLEN && idx1 >= 0 && idx1 < L_) ? inv_count : 0.0f;

            // B-matrix 4x16: same K striping, N = col. Clamped rows keep
            // loads in-bounds; invalid rows are zeroed by the weights.
            const int r0 = min(max(idx0, 0), L_ - 1);
            const int r1 = min(max(idx1, 0), L_ - 1);
            const size_t o0 = (size_t)r0 * D_ + (n0 + col);
            const size_t o1 = (size_t)r1 * D_ + (n0 + col);
            v2f bm;
            bm.x = altb[o0] - refb[o0];
            bm.y = altb[o1] - refb[o1];

            // 8 args: (neg_a, A, neg_b, B, c_mod, C, reuse_a, reuse_b)
            acc = __builtin_amdgcn_wmma_f32_16x16x4_f32(
                false, a, false, bm, (short)0, acc, false, false);
        }
        // D layout: lanes 0-15, VGPR0 = (M=0, N=col); all rows identical.
        if (half == 0) s_pooled[n0 + col] = acc[0];
    }
    __syncthreads();

    // ---- LayerNorm over D=768 (3 channels per thread) ----
    const float x0 = s_pooled[tid];
    const float x1 = s_pooled[tid + 256];
    const float x2 = s_pooled[tid + 512];
    float s  = x0 + x1 + x2;
    float ss = x0 * x0 + x1 * x1 + x2 * x2;
    #pragma unroll
    for (int off = 16; off >= 1; off >>= 1) {
        s  += __shfl_xor(s,  off, 32);
        ss += __shfl_xor(ss, off, 32);
    }
    if (lane == 0) { s_part[wave] = s; s_part[8 + wave] = ss; }
    __syncthreads();

    float S = 0.0f, SS = 0.0f;
    #pragma unroll
    for (int i = 0; i < 8; ++i) { S += s_part[i]; SS += s_part[8 + i]; }
    const float mu   = S * (1.0f / D_);
    const float var  = SS * (1.0f / D_) - mu * mu;   // biased variance
    const float rsig = rsqrtf(var + LN_EPS);

    float* __restrict__ outb = out + (size_t)b * D_;
    outb[tid]       = (x0 - mu) * rsig * gamma[tid]       + beta[tid];
    outb[tid + 256] = (x1 - mu) * rsig * gamma[tid + 256] + beta[tid + 256];
    outb[tid + 512] = (x2 - mu) * rsig * gamma[tid + 512] + beta[tid + 512];
}

extern "C" void kernel_launch(void* const* d_in, const int* in_sizes, int n_in,
                              void* d_out, int out_size, void* d_ws, size_t ws_size,
                              hipStream_t stream) {
    (void)in_sizes; (void)n_in; (void)out_size; (void)d_ws; (void)ws_size;
    const float* ref_repr    = (const float*)d_in[0];
    const float* alt_repr    = (const float*)d_in[1];
    const int*   variant_pos = (const int*)  d_in[2];
    const float* gamma       = (const float*)d_in[3];
    const float* beta        = (const float*)d_in[4];
    float*       out         = (float*)d_out;

    vdp_wmma_kernel<<<dim3(B_), dim3(256), 0, stream>>>(
        ref_repr, alt_repr, variant_pos, gamma, beta, out);
}